// GraphGATConv_51539608129
// MI455X (gfx1250) — compile-verified
//
#include <hip/hip_runtime.h>
#include <math.h>

// Problem constants (from reference)
#define NN 50000
#define EE 800000
#define IND 128
#define EDD 32
#define HH 4
#define CC 16
#define HCC 64
#define GG 64
#define NCC 2
#define NEG_SLOPE 0.2f
#define LN_EPS 1e-5f

typedef __attribute__((ext_vector_type(2))) float v2f;
typedef __attribute__((ext_vector_type(8))) float v8f;

// ---------- small utility kernels ----------

__global__ void fill_kernel(float* p, int n, float v) {
  int i = blockIdx.x * blockDim.x + threadIdx.x;
  if (i < n) p[i] = v;
}

__global__ void bias_init_kernel(float* out, const float* b, int n) {
  int i = blockIdx.x * blockDim.x + threadIdx.x;
  if (i < n) out[i] = b[i & (HCC - 1)];
}

// ---------- self-loop edge_attr mean ----------

__global__ void loop_accum_kernel(const float* ea, const int* dst,
                                  float* loopea, float* cnt, int n /*E*32*/) {
  int i = blockIdx.x * blockDim.x + threadIdx.x;
  if (i >= n) return;
  int e = i >> 5, ch = i & 31;
  int d = dst[e];
  atomicAdd(&loopea[(size_t)d * EDD + ch], ea[i]);
  if (ch == 0) atomicAdd(&cnt[d], 1.0f);
}

__global__ void loop_div_kernel(float* loopea, const float* cnt, int n /*N*32*/) {
  int i = blockIdx.x * blockDim.x + threadIdx.x;
  if (i >= n) return;
  loopea[i] /= fmaxf(cnt[i >> 5], 1.0f);
}

// ---------- WMMA fp32 GEMM: C[M,Ncols] = A[M,K] @ B[K,Ncols], Ncols==64 ----------
// One wave computes a 16x16 tile via V_WMMA_F32_16X16X4_F32; 4 waves/block cover 64 cols.
// f32 A 16x4 layout (ISA 7.12.2): lanes 0-15 hold K=0(v0),K=1(v1); lanes 16-31 K=2,K=3.
// f32 C/D: VGPR j -> row j (lanes 0-15) / row j+8 (lanes 16-31), col = lane%16.

__global__ __launch_bounds__(128) void gemm_wmma_kernel(const float* __restrict__ A,
                                                        const float* __restrict__ B,
                                                        float* __restrict__ C,
                                                        int M, int K) {
  const int Ncols = HCC;
  int wave = threadIdx.x >> 5;
  int lane = threadIdx.x & 31;
  int half = lane >> 4;      // K-half selector
  int l = lane & 15;         // M (for A) or N (for B) index
  int row0 = blockIdx.x * 16;
  int col0 = wave * 16;
  if (row0 >= M) return;

  v8f acc = {};
  for (int k = 0; k < K; k += 4) {
    v2f a, b;
    const float* ap = A + (size_t)(row0 + l) * K + k + half * 2;
    a.x = ap[0];
    a.y = ap[1];
    const float* bp = B + (size_t)(k + half * 2) * Ncols + col0 + l;
    b.x = bp[0];
    b.y = bp[Ncols];
    acc = __builtin_amdgcn_wmma_f32_16x16x4_f32(false, a, false, b,
                                                (short)0, acc, false, false);
  }
  float* cp = C + (size_t)(row0 + half * 8) * Ncols + col0 + l;
#pragma unroll
  for (int j = 0; j < 8; ++j) cp[(size_t)j * Ncols] = acc[j];
}

// ---------- attention scalar precomputes ----------

// WeA[k][h] = sum_c We[k, h*16+c] * ae[h*16+c]   (collapses edge GEMM)
__global__ void wea_kernel(const float* We, const float* ae, float* weA) {
  int i = threadIdx.x;
  if (i < EDD * HH) {
    int k = i >> 2, h = i & 3;
    float s = 0.f;
#pragma unroll
    for (int c = 0; c < CC; ++c) s += We[k * HCC + h * CC + c] * ae[h * CC + c];
    weA[k * HH + h] = s;
  }
}

// al_src / al_dst per (node, head)
__global__ void alsd_kernel(const float* xw, const float* as, const float* ad,
                            float* alsrc, float* aldst, int n /*N*H*/) {
  int i = blockIdx.x * blockDim.x + threadIdx.x;
  if (i >= n) return;
  int node = i >> 2, h = i & 3;
  const float* xp = xw + (size_t)node * HCC + h * CC;
  float s1 = 0.f, s2 = 0.f;
#pragma unroll
  for (int c = 0; c < CC; ++c) {
    float v = xp[c];
    s1 += v * as[h * CC + c];
    s2 += v * ad[h * CC + c];
  }
  alsrc[i] = s1;
  aldst[i] = s2;
}

// al_e per edge (real edges use edge_attr; trailing N virtual edges use loop_ea)
__global__ void ale_kernel(const float* ea, const float* loopea,
                           const float* weA, float* ale, int Ef) {
  __shared__ float sW[EDD * HH];
  if (threadIdx.x < EDD * HH) sW[threadIdx.x] = weA[threadIdx.x];
  __syncthreads();
  int e = blockIdx.x * blockDim.x + threadIdx.x;
  if (e >= Ef) return;
  const float* row = (e < EE) ? (ea + (size_t)e * EDD)
                              : (loopea + (size_t)(e - EE) * EDD);
  float s0 = 0.f, s1 = 0.f, s2 = 0.f, s3 = 0.f;
#pragma unroll
  for (int k = 0; k < EDD; ++k) {
    float v = row[k];
    s0 += v * sW[k * 4 + 0];
    s1 += v * sW[k * 4 + 1];
    s2 += v * sW[k * 4 + 2];
    s3 += v * sW[k * 4 + 3];
  }
  float* o = ale + (size_t)e * 4;
  o[0] = s0; o[1] = s1; o[2] = s2; o[3] = s3;
}

// ---------- segment softmax (monotone uint mapping for float atomicMax) ----------

__device__ __forceinline__ unsigned fmap(float f) {
  unsigned u = __float_as_uint(f);
  return (u >> 31) ? ~u : (u | 0x80000000u);
}
__device__ __forceinline__ float funmap(unsigned u) {
  return (u >> 31) ? __uint_as_float(u & 0x7FFFFFFFu) : __uint_as_float(~u);
}
__device__ __forceinline__ void edge_sd(const int* src, const int* dst, int e,
                                        int& s, int& d) {
  if (e < EE) { s = src[e]; d = dst[e]; } else { s = d = e - EE; }
}

__global__ void passA_kernel(const int* src, const int* dst, const float* alsrc,
                             const float* aldst, float* ale, unsigned* amaxu,
                             int n /*Ef*4*/) {
  int idx = blockIdx.x * blockDim.x + threadIdx.x;
  if (idx >= n) return;
  int e = idx >> 2, h = idx & 3;
  int s, d;
  edge_sd(src, dst, e, s, d);
  float a = ale[idx] + alsrc[s * 4 + h] + aldst[d * 4 + h];
  a = (a > 0.f) ? a : NEG_SLOPE * a;  // leaky relu
  ale[idx] = a;
  atomicMax(&amaxu[d * 4 + h], fmap(a));
}

__global__ void passB_kernel(const int* src, const int* dst, float* ale,
                             const unsigned* amaxu, float* denom, int n) {
  int idx = blockIdx.x * blockDim.x + threadIdx.x;
  if (idx >= n) return;
  int e = idx >> 2, h = idx & 3;
  int s, d;
  edge_sd(src, dst, e, s, d);
  float ex = expf(ale[idx] - funmap(amaxu[d * 4 + h]));
  ale[idx] = ex;
  atomicAdd(&denom[d * 4 + h], ex);
}

__global__ void passC_kernel(const int* src, const int* dst, const float* ale,
                             const float* denom, const float* xw, float* out,
                             int n /*Ef*64*/) {
  int idx = blockIdx.x * blockDim.x + threadIdx.x;
  if (idx >= n) return;
  int e = idx >> 6, ch = idx & 63, h = ch >> 4;
  int s, d;
  edge_sd(src, dst, e, s, d);
  float att = ale[e * 4 + h] / (denom[d * 4 + h] + 1e-16f);
  atomicAdd(&out[(size_t)d * HCC + ch], att * xw[(size_t)s * HCC + ch]);
}

// ---------- relu + layernorm over 64 channels (one wave32 per node) ----------

__global__ __launch_bounds__(128) void relu_ln_kernel(float* h, const float* g,
                                                      const float* be, int n) {
  int wave = threadIdx.x >> 5, lane = threadIdx.x & 31;
  int node = blockIdx.x * 4 + wave;
  if (node >= n) return;
  float* p = h + (size_t)node * HCC;
  float v0 = fmaxf(p[lane], 0.f);
  float v1 = fmaxf(p[lane + 32], 0.f);
  float s = v0 + v1, sq = v0 * v0 + v1 * v1;
  for (int m = 16; m > 0; m >>= 1) {
    s += __shfl_xor(s, m, 32);
    sq += __shfl_xor(sq, m, 32);
  }
  float mu = s * (1.f / 64.f);
  float var = sq * (1.f / 64.f) - mu * mu;
  float inv = rsqrtf(var + LN_EPS);
  p[lane] = (v0 - mu) * inv * g[lane] + be[lane];
  p[lane + 32] = (v1 - mu) * inv * g[lane + 32] + be[lane + 32];
}

// ---------- relu fused into global max pool (non-negative => uint compare ok) ----------

__global__ void pool_kernel(const float* h, const int* batch, unsigned* pooled,
                            int n /*N*64*/) {
  int i = blockIdx.x * blockDim.x + threadIdx.x;
  if (i >= n) return;
  int node = i >> 6, ch = i & 63;
  float v = fmaxf(h[i], 0.f);
  atomicMax(&pooled[batch[node] * HCC + ch], __float_as_uint(v));
}

// ---------- MLP head: 64->32, LN, relu, 32->2, log_softmax ----------

__global__ __launch_bounds__(32) void head_kernel(const float* pooled,
                                                  const float* fw1, const float* fb1,
                                                  const float* g2, const float* be2,
                                                  const float* fw2, const float* fb2,
                                                  float* out) {
  int g = blockIdx.x, t = threadIdx.x;  // t in [0,32)
  const float* p = pooled + (size_t)g * HCC;
  float z = fb1[t];
  for (int k = 0; k < HCC; ++k) z += p[k] * fw1[k * 32 + t];
  float s = z;
  for (int m = 16; m > 0; m >>= 1) s += __shfl_xor(s, m, 32);
  float mu = s * (1.f / 32.f);
  float d0 = z - mu;
  float sq = d0 * d0;
  for (int m = 16; m > 0; m >>= 1) sq += __shfl_xor(sq, m, 32);
  float inv = rsqrtf(sq * (1.f / 32.f) + LN_EPS);
  float zn = fmaxf(d0 * inv * g2[t] + be2[t], 0.f);
  float l0 = zn * fw2[t * 2 + 0];
  float l1 = zn * fw2[t * 2 + 1];
  for (int m = 16; m > 0; m >>= 1) {
    l0 += __shfl_xor(l0, m, 32);
    l1 += __shfl_xor(l1, m, 32);
  }
  if (t == 0) {
    l0 += fb2[0];
    l1 += fb2[1];
    float mx = fmaxf(l0, l1);
    float lse = mx + logf(expf(l0 - mx) + expf(l1 - mx));
    out[g * 2 + 0] = l0 - lse;
    out[g * 2 + 1] = l1 - lse;
  }
}

// ---------- host orchestration ----------

static inline int cdiv(long long a, int b) { return (int)((a + b - 1) / b); }

extern "C" void kernel_launch(void* const* d_in, const int* in_sizes, int n_in,
                              void* d_out, int out_size, void* d_ws, size_t ws_size,
                              hipStream_t stream) {
  (void)in_sizes; (void)n_in; (void)out_size; (void)ws_size;
  const float* x   = (const float*)d_in[0];
  const float* ea  = (const float*)d_in[1];
  const float* W1  = (const float*)d_in[2];
  const float* as1 = (const float*)d_in[3];
  const float* ad1 = (const float*)d_in[4];
  const float* We1 = (const float*)d_in[5];
  const float* ae1 = (const float*)d_in[6];
  const float* b1  = (const float*)d_in[7];
  const float* W2  = (const float*)d_in[8];
  const float* as2 = (const float*)d_in[9];
  const float* ad2 = (const float*)d_in[10];
  const float* We2 = (const float*)d_in[11];
  const float* ae2 = (const float*)d_in[12];
  const float* b2  = (const float*)d_in[13];
  const float* g1  = (const float*)d_in[14];
  const float* be1 = (const float*)d_in[15];
  const float* fw1 = (const float*)d_in[16];
  const float* fb1 = (const float*)d_in[17];
  const float* g2  = (const float*)d_in[18];
  const float* be2 = (const float*)d_in[19];
  const float* fw2 = (const float*)d_in[20];
  const float* fb2 = (const float*)d_in[21];
  const int* ei    = (const int*)d_in[22];
  const int* batch = (const int*)d_in[23];
  const int* srcp = ei;
  const int* dstp = ei + EE;
  float* out = (float*)d_out;

  const int Ef = EE + NN;

  // workspace carve-up (floats)
  float* ws = (float*)d_ws;
  float* xw     = ws;                          // N*64
  float* hbuf   = xw + (size_t)NN * HCC;       // N*64
  float* alsrc  = hbuf + (size_t)NN * HCC;     // N*4
  float* aldst  = alsrc + (size_t)NN * HH;     // N*4
  float* amaxf  = aldst + (size_t)NN * HH;     // N*4 (uint-mapped)
  float* denom  = amaxf + (size_t)NN * HH;     // N*4
  float* loopea = denom + (size_t)NN * HH;     // N*32
  float* cnt    = loopea + (size_t)NN * EDD;   // N
  float* ale    = cnt + (size_t)NN;            // (E+N)*4
  float* weA    = ale + (size_t)Ef * HH;       // 32*4
  float* pooled = weA + EDD * HH;              // G*64
  unsigned* amaxu   = (unsigned*)amaxf;
  unsigned* pooledu = (unsigned*)pooled;

  const int TB = 256;

  // self-loop edge_attr mean
  fill_kernel<<<cdiv(NN, TB), TB, 0, stream>>>(cnt, NN, 0.f);
  fill_kernel<<<cdiv((long long)NN * EDD, TB), TB, 0, stream>>>(loopea, NN * EDD, 0.f);
  loop_accum_kernel<<<cdiv((long long)EE * EDD, TB), TB, 0, stream>>>(ea, dstp, loopea, cnt, EE * EDD);
  loop_div_kernel<<<cdiv((long long)NN * EDD, TB), TB, 0, stream>>>(loopea, cnt, NN * EDD);

  // ===== GAT layer 1 =====
  gemm_wmma_kernel<<<NN / 16, 128, 0, stream>>>(x, W1, xw, NN, IND);
  wea_kernel<<<1, 128, 0, stream>>>(We1, ae1, weA);
  alsd_kernel<<<cdiv((long long)NN * HH, TB), TB, 0, stream>>>(xw, as1, ad1, alsrc, aldst, NN * HH);
  ale_kernel<<<cdiv(Ef, TB), TB, 0, stream>>>(ea, loopea, weA, ale, Ef);
  fill_kernel<<<cdiv((long long)NN * HH, TB), TB, 0, stream>>>(amaxf, NN * HH, 0.f);  // uint 0 == lowest mapped
  fill_kernel<<<cdiv((long long)NN * HH, TB), TB, 0, stream>>>(denom, NN * HH, 0.f);
  bias_init_kernel<<<cdiv((long long)NN * HCC, TB), TB, 0, stream>>>(hbuf, b1, NN * HCC);
  passA_kernel<<<cdiv((long long)Ef * HH, TB), TB, 0, stream>>>(srcp, dstp, alsrc, aldst, ale, amaxu, Ef * HH);
  passB_kernel<<<cdiv((long long)Ef * HH, TB), TB, 0, stream>>>(srcp, dstp, ale, amaxu, denom, Ef * HH);
  passC_kernel<<<cdiv((long long)Ef * HCC, TB), TB, 0, stream>>>(srcp, dstp, ale, denom, xw, hbuf, Ef * HCC);
  relu_ln_kernel<<<cdiv(NN, 4), 128, 0, stream>>>(hbuf, g1, be1, NN);

  // ===== GAT layer 2 =====
  gemm_wmma_kernel<<<NN / 16, 128, 0, stream>>>(hbuf, W2, xw, NN, HCC);
  wea_kernel<<<1, 128, 0, stream>>>(We2, ae2, weA);
  alsd_kernel<<<cdiv((long long)NN * HH, TB), TB, 0, stream>>>(xw, as2, ad2, alsrc, aldst, NN * HH);
  ale_kernel<<<cdiv(Ef, TB), TB, 0, stream>>>(ea, loopea, weA, ale, Ef);
  fill_kernel<<<cdiv((long long)NN * HH, TB), TB, 0, stream>>>(amaxf, NN * HH, 0.f);
  fill_kernel<<<cdiv((long long)NN * HH, TB), TB, 0, stream>>>(denom, NN * HH, 0.f);
  bias_init_kernel<<<cdiv((long long)NN * HCC, TB), TB, 0, stream>>>(hbuf, b2, NN * HCC);
  passA_kernel<<<cdiv((long long)Ef * HH, TB), TB, 0, stream>>>(srcp, dstp, alsrc, aldst, ale, amaxu, Ef * HH);
  passB_kernel<<<cdiv((long long)Ef * HH, TB), TB, 0, stream>>>(srcp, dstp, ale, amaxu, denom, Ef * HH);
  passC_kernel<<<cdiv((long long)Ef * HCC, TB), TB, 0, stream>>>(srcp, dstp, ale, denom, xw, hbuf, Ef * HCC);

  // relu fused into max pool; pooled init 0 == relu floor and isfinite fixup
  fill_kernel<<<cdiv((long long)GG * HCC, TB), TB, 0, stream>>>(pooled, GG * HCC, 0.f);
  pool_kernel<<<cdiv((long long)NN * HCC, TB), TB, 0, stream>>>(hbuf, batch, pooledu, NN * HCC);

  head_kernel<<<GG, 32, 0, stream>>>(pooled, fw1, fb1, g2, be2, fw2, fb2, out);
}